// Routing_52226802319486
// MI455X (gfx1250) — compile-verified
//
#include <hip/hip_runtime.h>
#include <math.h>

// EM routing (matrix capsules) for MI455X / gfx1250.
// Key decision: V (b,i,j,c,h) is 256MB fp32 -> never materialized in HBM.
// Instead V is recomputed per pass with V_WMMA_F32_16X16X4_F32; the inputs it
// needs (W 32MB + Bp 16MB + mu 8MB = 56MB) fit in the 192MB L2, so all 6
// recompute passes after the first run at L2 bandwidth instead of HBM.

#define B_SZ   16
#define N_INP  1024
#define N_OUT  32
#define D_SPC  8
#define D_INP  16
#define D_OUT  16
#define CH     (D_SPC * D_OUT)     // 128
#define EPS    1e-5f
#define IPB    32                  // input capsules per block in accum kernels

typedef __attribute__((ext_vector_type(2))) float v2f;
typedef __attribute__((ext_vector_type(8))) float v8f;

__device__ __forceinline__ float sigmoidf_(float x) { return 1.0f / (1.0f + __expf(-x)); }

__device__ __forceinline__ v8f wmma_f32_16x16x4(v2f a, v2f b, v8f c) {
  // 8 args: (neg_a, A, neg_b, B, c_mod, C, reuse_a, reuse_b)
  return __builtin_amdgcn_wmma_f32_16x16x4_f32(false, a, false, b, (short)0, c,
                                               false, false);
}

// ---------------------------------------------------------------------------
// Pass A: recompute V, compute R (softmax over j), accumulate D_use sums.
// Block: 256 threads (8 waves). Grid: (B/2, N_INP/IPB).
// WMMA tile: M=16 rows = (2 batches x 8 spatial c), N=16 = one j's 16 h, K=16 d.
// D-matrix layout: VGPR r, lane L -> row m = r + 8*(L>=16), col n = L&15
//   => b_local = L>=16, c = r, h = L&15, so per-lane ownership is clean.
// ---------------------------------------------------------------------------
__global__ __launch_bounds__(256) void accumA_kernel(
    const float* __restrict__ a_inp, const float* __restrict__ mu_inp,
    const float* __restrict__ W, const float* __restrict__ Bp,
    const float* __restrict__ beta_use, const float* __restrict__ beta_ign,
    const float* __restrict__ mu_prev,     // [B,N_OUT,CH] (prev iter mu_out)
    const float* __restrict__ inv2sig,     // [B,N_OUT,CH] 1/(2*sig2)
    const float* __restrict__ logdet,      // [B,N_OUT]    0.5*sum log sig2
    const float* __restrict__ logsig_a,    // [B,N_OUT]    log_sigmoid(a_out)
    float* __restrict__ Duse, float* __restrict__ mu_acc,
    float* __restrict__ denom, float* __restrict__ a_acc, int first) {
  __shared__ float sl[2][N_OUT];
  const int lane = threadIdx.x & 31;
  const int wave = threadIdx.x >> 5;
  const int half = lane >> 4;      // which 16-lane half
  const int l    = lane & 15;      // column (h) / A-row index
  const int b0   = blockIdx.x * 2;
  const int bD   = b0 + half;      // batch owned by this lane for C/D
  const int bA   = b0 + (l >> 3);  // batch of the A row this lane loads
  const int cA   = l & 7;
  const int i0   = blockIdx.y * IPB;
  const int jb   = wave * 4;       // this wave handles j = jb..jb+3
  const int kofs = half * 2;       // K offset inside a 4-wide K step

  float muAcc[4][8], dAcc[4], aAcc[4];
  float i2s[4][8], mpv[4][8], ldet[4], lsa[4];
  for (int t = 0; t < 4; t++) {
    dAcc[t] = 0.f; aAcc[t] = 0.f;
    for (int r = 0; r < 8; r++) muAcc[t][r] = 0.f;
  }
  if (!first) {
    for (int t = 0; t < 4; t++) {
      int j = jb + t;
      ldet[t] = logdet[bD * N_OUT + j];
      lsa[t]  = logsig_a[bD * N_OUT + j];
      for (int r = 0; r < 8; r++) {
        int idx = ((bD * N_OUT + j) * D_SPC + r) * D_OUT + l;
        i2s[t][r] = inv2sig[idx];
        mpv[t][r] = mu_prev[idx];
      }
    }
  }

  for (int ii = 0; ii < IPB; ii++) {
    int i = i0 + ii;
    // A fragments: A[m=l][k] = mu[bA, i, cA, k]; ISA 32-bit 16x4 layout.
    v2f Af[4];
    const float* mrow = mu_inp + ((size_t)(bA * N_INP + i) * D_SPC + cA) * D_INP;
#pragma unroll
    for (int kb = 0; kb < 4; kb++) {
      Af[kb].x = mrow[kb * 4 + kofs + 0];
      Af[kb].y = mrow[kb * 4 + kofs + 1];
    }
    float fa = sigmoidf_(a_inp[bD * N_INP + i]);

    float v[4][8];
#pragma unroll
    for (int t = 0; t < 4; t++) {
      int j = jb + t;
      const float* wrow = W + (size_t)(i * N_OUT + j) * D_INP * D_OUT;  // [d][h]
      v8f acc = {0.f, 0.f, 0.f, 0.f, 0.f, 0.f, 0.f, 0.f};
#pragma unroll
      for (int kb = 0; kb < 4; kb++) {
        v2f Bf;
        Bf.x = wrow[(kb * 4 + kofs + 0) * D_OUT + l];
        Bf.y = wrow[(kb * 4 + kofs + 1) * D_OUT + l];
        acc = wmma_f32_16x16x4(Af[kb], Bf, acc);
      }
      const float* bprow = Bp + (size_t)(i * N_OUT + j) * CH;
#pragma unroll
      for (int r = 0; r < 8; r++) v[t][r] = acc[r] + bprow[r * D_OUT + l];
    }

    float R[4];
    if (first) {
#pragma unroll
      for (int t = 0; t < 4; t++) R[t] = 1.0f / (float)N_OUT;
    } else {
      // log_p = -sum_ch (V-mu)^2/(2 sig2) - 0.5*sum log sig2
#pragma unroll
      for (int t = 0; t < 4; t++) {
        float s = 0.f;
#pragma unroll
        for (int r = 0; r < 8; r++) {
          float d = v[t][r] - mpv[t][r];
          s += d * d * i2s[t][r];
        }
#pragma unroll
        for (int m = 1; m < 16; m <<= 1) s += __shfl_xor(s, m, 32);  // over h
        if (l == 0) sl[half][jb + t] = lsa[t] - s - ldet[t];
      }
      __syncthreads();
      float mx = -3.4e38f;
      for (int jj = 0; jj < N_OUT; jj++) mx = fmaxf(mx, sl[half][jj]);
      float sum = 0.f;
      for (int jj = 0; jj < N_OUT; jj++) sum += __expf(sl[half][jj] - mx);
      float inv = 1.0f / sum;
#pragma unroll
      for (int t = 0; t < 4; t++) R[t] = __expf(sl[half][jb + t] - mx) * inv;
      __syncthreads();
    }

#pragma unroll
    for (int t = 0; t < 4; t++) {
      float du = fa * R[t];
      if (l == 0) {  // one lane per (b_local, j)
        int j = jb + t;
        Duse[((size_t)bD * N_INP + i) * N_OUT + j] = du;
        dAcc[t] += du;
        aAcc[t] += (beta_use[i * N_OUT + j] + beta_ign[i * N_OUT + j]) * du;
      }
#pragma unroll
      for (int r = 0; r < 8; r++) muAcc[t][r] += du * v[t][r];
    }
  }

  // flush register partials (per-block) with fp32 global atomics
#pragma unroll
  for (int t = 0; t < 4; t++) {
    int j = jb + t;
#pragma unroll
    for (int r = 0; r < 8; r++)
      atomicAdd(&mu_acc[((bD * N_OUT + j) * D_SPC + r) * D_OUT + l], muAcc[t][r]);
    if (l == 0) {
      atomicAdd(&denom[bD * N_OUT + j], dAcc[t]);
      atomicAdd(&a_acc[bD * N_OUT + j], aAcc[t]);
    }
  }
}

// ---------------------------------------------------------------------------
// Pass B: recompute V (WMMA again, hits L2), accumulate D_use*(V-mu_out)^2.
// ---------------------------------------------------------------------------
__global__ __launch_bounds__(256) void accumB_kernel(
    const float* __restrict__ mu_inp, const float* __restrict__ W,
    const float* __restrict__ Bp, const float* __restrict__ Duse,
    const float* __restrict__ mu_out, float* __restrict__ sig_acc) {
  const int lane = threadIdx.x & 31;
  const int wave = threadIdx.x >> 5;
  const int half = lane >> 4;
  const int l    = lane & 15;
  const int b0   = blockIdx.x * 2;
  const int bD   = b0 + half;
  const int bA   = b0 + (l >> 3);
  const int cA   = l & 7;
  const int i0   = blockIdx.y * IPB;
  const int jb   = wave * 4;
  const int kofs = half * 2;

  float sg[4][8], mo[4][8];
  for (int t = 0; t < 4; t++)
    for (int r = 0; r < 8; r++) {
      sg[t][r] = 0.f;
      mo[t][r] = mu_out[((bD * N_OUT + jb + t) * D_SPC + r) * D_OUT + l];
    }

  for (int ii = 0; ii < IPB; ii++) {
    int i = i0 + ii;
    v2f Af[4];
    const float* mrow = mu_inp + ((size_t)(bA * N_INP + i) * D_SPC + cA) * D_INP;
#pragma unroll
    for (int kb = 0; kb < 4; kb++) {
      Af[kb].x = mrow[kb * 4 + kofs + 0];
      Af[kb].y = mrow[kb * 4 + kofs + 1];
    }
#pragma unroll
    for (int t = 0; t < 4; t++) {
      int j = jb + t;
      const float* wrow = W + (size_t)(i * N_OUT + j) * D_INP * D_OUT;
      v8f acc = {0.f, 0.f, 0.f, 0.f, 0.f, 0.f, 0.f, 0.f};
#pragma unroll
      for (int kb = 0; kb < 4; kb++) {
        v2f Bf;
        Bf.x = wrow[(kb * 4 + kofs + 0) * D_OUT + l];
        Bf.y = wrow[(kb * 4 + kofs + 1) * D_OUT + l];
        acc = wmma_f32_16x16x4(Af[kb], Bf, acc);
      }
      float du = Duse[((size_t)bD * N_INP + i) * N_OUT + j];
      const float* bprow = Bp + (size_t)(i * N_OUT + j) * CH;
#pragma unroll
      for (int r = 0; r < 8; r++) {
        float vv = acc[r] + bprow[r * D_OUT + l];
        float d = vv - mo[t][r];
        sg[t][r] += du * d * d;
      }
    }
  }
#pragma unroll
  for (int t = 0; t < 4; t++)
#pragma unroll
    for (int r = 0; r < 8; r++)
      atomicAdd(&sig_acc[((bD * N_OUT + jb + t) * D_SPC + r) * D_OUT + l],
                sg[t][r]);
}

// finalize A: mu_out = mu_acc/denom, a_out; zero sig_acc for pass B
__global__ __launch_bounds__(128) void finalizeA_kernel(
    const float* __restrict__ mu_acc, const float* __restrict__ denom,
    const float* __restrict__ a_acc, const float* __restrict__ bias_ign_fa,
    float* __restrict__ out_a, float* __restrict__ out_mu,
    float* __restrict__ sig_acc) {
  int bj = blockIdx.x, t = threadIdx.x;
  float over = 1.0f / (denom[bj] + EPS);
  out_mu[bj * CH + t] = mu_acc[bj * CH + t] * over;
  sig_acc[bj * CH + t] = 0.f;
  if (t == 0) out_a[bj] = a_acc[bj] - bias_ign_fa[bj];
}

// finalize B: sig2, 1/(2 sig2), 0.5*sum log sig2, log_sigmoid(a_out);
// re-zero the pass-A accumulators for the next iteration.
__global__ __launch_bounds__(128) void finalizeB_kernel(
    const float* sig_acc, float* denom, const float* out_a, float* out_sig,
    float* inv2sig, float* logdet, float* logsig_a, float* mu_acc,
    float* a_acc) {
  __shared__ float red[128];
  int bj = blockIdx.x, t = threadIdx.x;
  float over = 1.0f / (denom[bj] + EPS);
  float s2 = sig_acc[bj * CH + t] * over + EPS;
  out_sig[bj * CH + t] = s2;
  inv2sig[bj * CH + t] = 1.0f / (2.0f * s2);
  red[t] = 0.5f * __logf(s2);
  __syncthreads();
  for (int s = 64; s > 0; s >>= 1) {
    if (t < s) red[t] += red[t + s];
    __syncthreads();
  }
  mu_acc[bj * CH + t] = 0.f;
  if (t == 0) {
    logdet[bj] = red[0];
    float a = out_a[bj];
    logsig_a[bj] = (a >= 0.f) ? -log1pf(__expf(-a)) : (a - log1pf(__expf(a)));
    denom[bj] = 0.f;
    a_acc[bj] = 0.f;
  }
}

// iteration-invariant: sum_i beta_ign[i,j]*sigmoid(a_inp[b,i])
__global__ __launch_bounds__(256) void precompute_kernel(
    const float* __restrict__ a_inp, const float* __restrict__ beta_ign,
    float* __restrict__ bias_ign_fa) {
  __shared__ float red[256];
  int bj = blockIdx.x, b = bj >> 5, j = bj & (N_OUT - 1);
  float s = 0.f;
  for (int i = threadIdx.x; i < N_INP; i += 256)
    s += beta_ign[i * N_OUT + j] * sigmoidf_(a_inp[b * N_INP + i]);
  red[threadIdx.x] = s;
  __syncthreads();
  for (int k = 128; k > 0; k >>= 1) {
    if (threadIdx.x < k) red[threadIdx.x] += red[threadIdx.x + k];
    __syncthreads();
  }
  if (threadIdx.x == 0) bias_ign_fa[bj] = red[0];
}

__global__ void zero_kernel(float* p, int n) {
  int i = blockIdx.x * 256 + threadIdx.x;
  if (i < n) p[i] = 0.f;
}

extern "C" void kernel_launch(void* const* d_in, const int* in_sizes, int n_in,
                              void* d_out, int out_size, void* d_ws,
                              size_t ws_size, hipStream_t stream) {
  const float* a_inp    = (const float*)d_in[0];
  const float* mu_inp   = (const float*)d_in[1];
  const float* W        = (const float*)d_in[2];
  const float* Bp       = (const float*)d_in[3];
  const float* beta_use = (const float*)d_in[4];
  const float* beta_ign = (const float*)d_in[5];

  float* out     = (float*)d_out;
  float* out_a   = out;                          // [B, N_OUT]
  float* out_mu  = out + B_SZ * N_OUT;           // [B, N_OUT, CH]
  float* out_sig = out_mu + B_SZ * N_OUT * CH;   // [B, N_OUT, CH]

  // workspace layout (floats), ~2.9 MB total
  float* ws          = (float*)d_ws;
  float* Duse        = ws;                                   // B*N_INP*N_OUT
  float* mu_acc      = Duse + (size_t)B_SZ * N_INP * N_OUT;  // B*N_OUT*CH
  float* sig_acc     = mu_acc + B_SZ * N_OUT * CH;           // B*N_OUT*CH
  float* denom       = sig_acc + B_SZ * N_OUT * CH;          // B*N_OUT
  float* a_acc       = denom + B_SZ * N_OUT;                 // B*N_OUT
  float* bias_ign_fa = a_acc + B_SZ * N_OUT;                 // B*N_OUT
  float* inv2sig     = bias_ign_fa + B_SZ * N_OUT;           // B*N_OUT*CH
  float* logdet      = inv2sig + B_SZ * N_OUT * CH;          // B*N_OUT
  float* logsig_a    = logdet + B_SZ * N_OUT;                // B*N_OUT

  // zero the (contiguous) accumulators mu_acc..a_acc
  int nz = B_SZ * N_OUT * CH * 2 + B_SZ * N_OUT * 2;
  zero_kernel<<<(nz + 255) / 256, 256, 0, stream>>>(mu_acc, nz);
  precompute_kernel<<<B_SZ * N_OUT, 256, 0, stream>>>(a_inp, beta_ign,
                                                      bias_ign_fa);
  dim3 gA(B_SZ / 2, N_INP / IPB);
  for (int it = 0; it < 3; ++it) {
    accumA_kernel<<<gA, 256, 0, stream>>>(
        a_inp, mu_inp, W, Bp, beta_use, beta_ign, out_mu, inv2sig, logdet,
        logsig_a, Duse, mu_acc, denom, a_acc, it == 0 ? 1 : 0);
    finalizeA_kernel<<<B_SZ * N_OUT, 128, 0, stream>>>(
        mu_acc, denom, a_acc, bias_ign_fa, out_a, out_mu, sig_acc);
    accumB_kernel<<<gA, 256, 0, stream>>>(mu_inp, W, Bp, Duse, out_mu, sig_acc);
    finalizeB_kernel<<<B_SZ * N_OUT, 128, 0, stream>>>(
        sig_acc, denom, out_a, out_sig, inv2sig, logdet, logsig_a, mu_acc,
        a_acc);
  }
}